// FragmentPositionDistribution2_64802466562897
// MI455X (gfx1250) — compile-verified
//
#include <hip/hip_runtime.h>
#include <hip/hip_bf16.h>
#include <math.h>
#include <stdint.h>

#define BINSIZE   200
#define WIDTH_TOT 100000
#define NBIN      500                     // BINWIDTH = WIDTH/BINSIZE
#define LOG_BINSIZE 5.2983173665480363f   // log(200)

// ---- CDNA5 async global->LDS path (guarded; falls back to plain loads) ----
#if defined(__has_builtin)
#  if __has_builtin(__builtin_amdgcn_global_load_async_to_lds_b32) && \
      __has_builtin(__builtin_amdgcn_s_wait_asynccnt)
#    define USE_ASYNC_LDS 1
#  endif
#endif
#ifndef USE_ASYNC_LDS
#  define USE_ASYNC_LDS 0
#endif

__device__ __forceinline__ float waveMax(float v) {
  #pragma unroll
  for (int o = 16; o >= 1; o >>= 1) v = fmaxf(v, __shfl_xor(v, o, 32));
  return v;
}
__device__ __forceinline__ float waveSum(float v) {
  #pragma unroll
  for (int o = 16; o >= 1; o >>= 1) v += __shfl_xor(v, o, 32);
  return v;
}

// Phase A: one block per (cluster = blockIdx.x, region_oi = blockIdx.y) row.
// full_mode=1: write full heights table [R_oi, C, NBIN] to out.
// full_mode=0: write only logsumexp Z [R_oi, C] to out.
__global__ __launch_bounds__(512)
void heights_kernel(const float* __restrict__ baseline,   // [n_regions, NBIN]
                    const float* __restrict__ delta,      // [n_regions, C, NBIN]
                    const int*   __restrict__ regions_oi, // [R_oi]
                    int n_clusters, int full_mode,
                    float* __restrict__ out) {
  const int c    = blockIdx.x;      // cluster
  const int r    = blockIdx.y;      // region-of-interest index
  const int reg  = regions_oi[r];
  const int b    = threadIdx.x;
  const int lane = threadIdx.x & 31;
  const int wid  = threadIdx.x >> 5;

  __shared__ float s_base[NBIN];
  __shared__ float sredA[16];
  __shared__ float sredB[16];
  __shared__ float sbc[2];

  const float* brow = baseline + (size_t)reg * NBIN;
#if USE_ASYNC_LDS
  typedef __attribute__((address_space(1))) int GINT;   // global AS
  typedef __attribute__((address_space(3))) int LINT;   // LDS AS
  if (b < NBIN) {
    __builtin_amdgcn_global_load_async_to_lds_b32(
        (GINT*)(uintptr_t)(brow + b),
        (LINT*)(uint32_t)(uintptr_t)(&s_base[b]),
        0, 0);
  }
  __builtin_amdgcn_s_wait_asynccnt(0);
  __syncthreads();
#else
  if (b < NBIN) s_base[b] = brow[b];
  __syncthreads();
#endif

  float u = -INFINITY;
  if (b < NBIN) {
    const float d = delta[((size_t)reg * n_clusters + c) * NBIN + b];
    u = s_base[b] + d;
  }

  // block max (16 wave32s)
  float m = waveMax(u);
  if (lane == 0) sredA[wid] = m;
  __syncthreads();
  if (wid == 0) {
    float t = (lane < 16) ? sredA[lane] : -INFINITY;
    t = waveMax(t);
    if (lane == 0) sbc[0] = t;
  }
  __syncthreads();
  m = sbc[0];

  // block sum of exp
  float e = (b < NBIN) ? expf(u - m) : 0.0f;
  float s = waveSum(e);
  if (lane == 0) sredB[wid] = s;
  __syncthreads();
  if (wid == 0) {
    float t = (lane < 16) ? sredB[lane] : 0.0f;
    t = waveSum(t);
    if (lane == 0) sbc[1] = t;
  }
  __syncthreads();

  const float lse = m + logf(sbc[1]);
  if (full_mode) {
    if (b < NBIN)
      out[((size_t)r * n_clusters + c) * NBIN + b] = u - lse - LOG_BINSIZE;
  } else {
    if (threadIdx.x == 0)
      out[(size_t)r * n_clusters + c] = lse;
  }
}

// Phase B: one thread per fragment.
__global__ __launch_bounds__(256)
void fragment_kernel(const float* __restrict__ tab,        // heights (full) or Z table
                     const float* __restrict__ baseline,
                     const float* __restrict__ delta,
                     const int*   __restrict__ regions_oi,
                     const int*   __restrict__ coords,     // [N,2]
                     const int*   __restrict__ lrix,       // [N]
                     const int*   __restrict__ lcix,       // [N]
                     const int*   __restrict__ labels,     // [n_cells]
                     const float* __restrict__ inside,     // [1]
                     int n, int n_clusters, int full_mode,
                     float* __restrict__ out) {
  const int i = blockIdx.x * blockDim.x + threadIdx.x;
  if (i >= n) return;

  const int2 cc = ((const int2*)coords)[i];          // global_load_b64
  const int bl = cc.x / BINSIZE;                     // coords >= 0 -> mul/shift
  const int br = cc.y / BINSIZE;
  const int r   = lrix[i];
  const int lab = labels[lcix[i]];

  float lp_left;
  if (full_mode) {
    lp_left = tab[((size_t)r * n_clusters + lab) * NBIN + bl];
  } else {
    const int reg = regions_oi[r];
    const float u = baseline[(size_t)reg * NBIN + bl]
                  + delta[((size_t)reg * n_clusters + lab) * NBIN + bl];
    lp_left = u - tab[(size_t)r * n_clusters + lab] - LOG_BINSIZE;
  }

  const float x  = inside[0];                        // uniform
  const float sg = 1.0f / (1.0f + expf(-x));
  const float li = logf(sg)        - logf((float)NBIN);               // log(500)
  const float lo = logf(1.0f - sg) - logf((float)(WIDTH_TOT - NBIN)); // log(99500)

  float2 o;
  o.x = lp_left;
  o.y = (bl == br) ? li : lo;
  ((float2*)out)[i] = o;                             // global_store_b64
}

extern "C" void kernel_launch(void* const* d_in, const int* in_sizes, int n_in,
                              void* d_out, int out_size, void* d_ws, size_t ws_size,
                              hipStream_t stream) {
  const float* baseline   = (const float*)d_in[0];
  const float* delta      = (const float*)d_in[1];
  const float* inside     = (const float*)d_in[2];
  const int*   regions_oi = (const int*)d_in[3];
  const int*   coords     = (const int*)d_in[4];
  const int*   lrix       = (const int*)d_in[5];
  const int*   lcix       = (const int*)d_in[6];
  const int*   labels     = (const int*)d_in[7];

  const int n_regions    = in_sizes[0] / NBIN;                 // 5000
  const int n_clusters   = in_sizes[1] / in_sizes[0];          // 20
  const int n_regions_oi = in_sizes[3];                        // 200
  const int n_frag       = in_sizes[4] / 2;                    // 2,000,000
  (void)n_regions;

  const size_t full_bytes = (size_t)n_regions_oi * n_clusters * NBIN * sizeof(float); // 8 MB
  const int full_mode = (ws_size >= full_bytes) ? 1 : 0;
  float* tab = (float*)d_ws;

  // Phase A: build heights (or Z) table in workspace. 2D grid avoids int div.
  dim3 gridA(n_clusters, n_regions_oi);   // 20 x 200 = 4000 blocks
  heights_kernel<<<gridA, 512, 0, stream>>>(baseline, delta, regions_oi,
                                            n_clusters, full_mode, tab);

  // Phase B: per-fragment gather.
  dim3 gridB((n_frag + 255) / 256);
  fragment_kernel<<<gridB, 256, 0, stream>>>(tab, baseline, delta, regions_oi,
                                             coords, lrix, lcix, labels, inside,
                                             n_frag, n_clusters, full_mode,
                                             (float*)d_out);
}